// MetricLoss_49890340110572
// MI455X (gfx1250) — compile-verified
//
#include <hip/hip_runtime.h>
#include <hip/hip_bf16.h>

#define K_LABELS 32
#define DIM      16
#define BATCH    4
#define IMG_H    384
#define IMG_W    384
#define NPIX     (IMG_H * IMG_W)        // 147456, divisible by 4
#define PUSH_MARGIN 0.25f
#define N_COMPARISONS 496.0f            // K*(K-1)/2 = 32*31/2
#define CENT_ELEMS (BATCH * K_LABELS * DIM)   // 2048 floats = 8KB

typedef float v2f __attribute__((ext_vector_type(2)));
typedef float v8f __attribute__((ext_vector_type(8)));
typedef unsigned int v4u __attribute__((ext_vector_type(4)));
typedef int v4i __attribute__((ext_vector_type(4)));
typedef int v8i __attribute__((ext_vector_type(8)));

// ---------------------------------------------------------------------------
// Kernel 1: per-batch centroid sums via one-hot WMMA segment-sum.
//   sums[b][k][d] = sum over pixels with label k of emb[b,p,d]
//   counts[b][k]  = #pixels with label k
// Each wave step consumes 4 consecutive pixels:
//   A (16x4 f32, one-hot): lanes 0-15 hold M=lane {K0,K1}; lanes 16-31 {K2,K3}
//   B (4x16 f32, emb rows): lanes 0-15 col N=lane K={0,1}; lanes 16-31 K={2,3}
//   C (16x16 f32): VGPR j -> M = j + 8*(lane>=16), N = lane&15
// ---------------------------------------------------------------------------
__global__ void centroid_kernel(const float* __restrict__ emb,
                                const int*   __restrict__ lab,
                                float* __restrict__ sums,    // [B][K][D]
                                float* __restrict__ counts)  // [B][K]
{
    const int b    = blockIdx.y;
    const int lane = threadIdx.x & 31;
    const int hf   = lane >> 4;          // lane-half: 0 -> K pixels {0,1}, 1 -> {2,3}
    const int m    = lane & 15;

    const int wavesPerBlock = blockDim.x >> 5;
    const int gwave  = blockIdx.x * wavesPerBlock + (threadIdx.x >> 5);
    const int nwaves = gridDim.x * wavesPerBlock;

    const float* e = emb + (size_t)b * NPIX * DIM;
    const int*   l = lab + (size_t)b * NPIX;

    v8f c0 = {};   // labels 0..15
    v8f c1 = {};   // labels 16..31
    float cnt = 0.0f;

    const int ngroups = NPIX / 4;
    #pragma unroll 2
    for (int g = gwave; g < ngroups; g += nwaves) {
        const int p = g * 4;

        const int4 l4 = *(const int4*)(l + p);

        // B operand: two coalesced f32 loads covering the 4x16 tile
        v2f bb;
        bb.x = e[(size_t)(p + 0 + 2 * hf) * DIM + m];
        bb.y = e[(size_t)(p + 1 + 2 * hf) * DIM + m];

        // labels feeding this lane-half's K slots
        const int la = hf ? l4.z : l4.x;
        const int lb = hf ? l4.w : l4.y;

        v2f a0, a1;
        a0.x = (la == m)        ? 1.0f : 0.0f;
        a0.y = (lb == m)        ? 1.0f : 0.0f;
        a1.x = (la == m + 16)   ? 1.0f : 0.0f;
        a1.y = (lb == m + 16)   ? 1.0f : 0.0f;

        c0 = __builtin_amdgcn_wmma_f32_16x16x4_f32(false, a0, false, bb,
                                                   (short)0, c0, false, false);
        c1 = __builtin_amdgcn_wmma_f32_16x16x4_f32(false, a1, false, bb,
                                                   (short)0, c1, false, false);

        // per-lane label counting: lane index == label id
        cnt += (float)((l4.x == lane) + (l4.y == lane) +
                       (l4.z == lane) + (l4.w == lane));
    }

    // Block-level reduction in LDS, then one set of global atomics per block.
    __shared__ float lsum[2 * 16 * 16];   // [group][M][N]
    __shared__ float lcnt[K_LABELS];
    for (int i = threadIdx.x; i < 512; i += blockDim.x) lsum[i] = 0.0f;
    if (threadIdx.x < K_LABELS) lcnt[threadIdx.x] = 0.0f;
    __syncthreads();

    #pragma unroll
    for (int j = 0; j < 8; ++j) {
        const int mi = j + 8 * hf;
        atomicAdd(&lsum[0 * 256 + mi * 16 + m], c0[j]);
        atomicAdd(&lsum[1 * 256 + mi * 16 + m], c1[j]);
    }
    atomicAdd(&lcnt[lane], cnt);
    __syncthreads();

    float* bsums = sums + (size_t)b * K_LABELS * DIM;
    for (int i = threadIdx.x; i < 512; i += blockDim.x)
        atomicAdd(&bsums[i], lsum[i]);
    if (threadIdx.x < K_LABELS)
        atomicAdd(&counts[b * K_LABELS + threadIdx.x], lcnt[threadIdx.x]);
}

// ---------------------------------------------------------------------------
// Kernel 2: cents = counts > 0 ? sums / max(counts,1) : 0
// ---------------------------------------------------------------------------
__global__ void finalize_cents(const float* __restrict__ sums,
                               const float* __restrict__ counts,
                               float* __restrict__ cents)
{
    const int i = blockIdx.x * blockDim.x + threadIdx.x;
    if (i >= CENT_ELEMS) return;
    const float c = counts[i / DIM];
    cents[i] = (c > 0.0f) ? sums[i] / fmaxf(c, 1.0f) : 0.0f;
}

// ---------------------------------------------------------------------------
// Kernel 3: push loss per batch (strict upper triangle of KxK L1 distances)
// ---------------------------------------------------------------------------
__global__ void push_kernel(const float* __restrict__ cents,
                            float* __restrict__ push)
{
    const int b = blockIdx.x;
    __shared__ float cl[K_LABELS * DIM];
    for (int i = threadIdx.x; i < K_LABELS * DIM; i += blockDim.x)
        cl[i] = cents[b * K_LABELS * DIM + i];
    __syncthreads();

    float acc = 0.0f;
    for (int pair = threadIdx.x; pair < K_LABELS * K_LABELS; pair += blockDim.x) {
        const int i = pair / K_LABELS;
        const int j = pair % K_LABELS;
        if (j > i) {
            float d = 0.0f;
            #pragma unroll
            for (int x = 0; x < DIM; ++x)
                d += fabsf(cl[i * DIM + x] - cl[j * DIM + x]);
            const float mrg = fmaxf(PUSH_MARGIN - d, 0.0f);
            acc += mrg * mrg;
        }
    }

    __shared__ float red[256];
    red[threadIdx.x] = acc;
    __syncthreads();
    for (int s = 128; s > 0; s >>= 1) {
        if ((int)threadIdx.x < s) red[threadIdx.x] += red[threadIdx.x + s];
        __syncthreads();
    }
    if (threadIdx.x == 0) push[b] = red[0] / N_COMPARISONS;
}

// ---------------------------------------------------------------------------
// Kernel 4: pull loss. For each centroid-batch bc, accumulate
//   sum over ALL pixels of (L1(emb_pixel, cents[bc][label]))^2
// Centroid table (8 KB) staged into LDS by the Tensor Data Mover: one
// tensor_load_to_lds (1 row x 2048 f32 tile) issued by wave 0, tracked with
// TENSORcnt, published to the workgroup via barrier. Grid-stride pixel loop,
// float4 loads (pass is L2-resident after the centroid pass).
// ---------------------------------------------------------------------------
__global__ void pull_kernel(const float* __restrict__ emb,
                            const int*   __restrict__ lab,
                            const float* __restrict__ cents,
                            float* __restrict__ pullacc)   // [B]
{
    // single LDS arena: cl at LDS offset 0 (required by the TDM descriptor)
    __shared__ float smem[CENT_ELEMS + 256];
    float* cl  = smem;              // [B][K][D]
    float* red = smem + CENT_ELEMS; // [256]

#if defined(__gfx1250__) && __has_builtin(__builtin_amdgcn_tensor_load_to_lds)
    if ((threadIdx.x >> 5) == 0) {   // wave 0 only (wave-uniform branch; TDM ignores EXEC)
        const unsigned long long ga = (unsigned long long)(uintptr_t)cents;
        // D# group 0: count=1 | lds_addr=0 | global_addr | type=2
        v4u g0 = { 1u,
                   0u,
                   (unsigned int)ga,
                   (unsigned int)((ga >> 32) & 0x01FFFFFFu) | 0x80000000u };
        // D# group 1: data_size=4B, tensor 2048x1, tile 2048x1, stride0=2048
        v8i g1 = { (int)(2u << 16),          // wg_mask=0 | data_size=2 (4B)
                   (int)(2048u << 16),       // tensor_dim0[15:0] @ bits[63:48]
                   (int)(0u | (1u << 16)),   // tensor_dim0[31:16]=0 | tensor_dim1[15:0]=1
                   (int)(2048u << 16),       // tensor_dim1[31:16]=0 | tile_dim0=2048
                   1,                        // tile_dim1=1 | tile_dim2=0
                   2048,                     // tensor_dim0_stride[31:0]
                   (int)(2048u << 16),       // stride0[47:32]=0 | tensor_dim1_stride[15:0]
                   0 };                      // tensor_dim1_stride[47:16]=0
        v4i gz  = { 0, 0, 0, 0 };            // groups 2/3 unused (2D tensor)
        v8i gz8 = { 0, 0, 0, 0, 0, 0, 0, 0 };
        __builtin_amdgcn_tensor_load_to_lds(g0, g1, gz, gz, gz8, 0);
        __builtin_amdgcn_s_wait_tensorcnt(0);
    }
    __syncthreads();
#else
    for (int i = threadIdx.x; i < CENT_ELEMS; i += blockDim.x) cl[i] = cents[i];
    __syncthreads();
#endif

    float acc[BATCH] = {0.0f, 0.0f, 0.0f, 0.0f};
    const int total  = BATCH * NPIX;
    const int stride = gridDim.x * blockDim.x;

    for (int p = blockIdx.x * blockDim.x + threadIdx.x; p < total; p += stride) {
        const int lbl = lab[p];
        const float4* e4 = (const float4*)(emb + (size_t)p * DIM);
        const float4 e0 = e4[0], e1 = e4[1], e2 = e4[2], e3 = e4[3];

        #pragma unroll
        for (int bc = 0; bc < BATCH; ++bc) {
            const float* cc = &cl[(bc * K_LABELS + lbl) * DIM];
            float d = fabsf(e0.x - cc[0])  + fabsf(e0.y - cc[1])
                    + fabsf(e0.z - cc[2])  + fabsf(e0.w - cc[3])
                    + fabsf(e1.x - cc[4])  + fabsf(e1.y - cc[5])
                    + fabsf(e1.z - cc[6])  + fabsf(e1.w - cc[7])
                    + fabsf(e2.x - cc[8])  + fabsf(e2.y - cc[9])
                    + fabsf(e2.z - cc[10]) + fabsf(e2.w - cc[11])
                    + fabsf(e3.x - cc[12]) + fabsf(e3.y - cc[13])
                    + fabsf(e3.z - cc[14]) + fabsf(e3.w - cc[15]);
            // relu(d - 0) == d since d >= 0
            acc[bc] += d * d;
        }
    }

    #pragma unroll
    for (int bc = 0; bc < BATCH; ++bc) {
        red[threadIdx.x] = acc[bc];
        __syncthreads();
        for (int s = 128; s > 0; s >>= 1) {
            if ((int)threadIdx.x < s) red[threadIdx.x] += red[threadIdx.x + s];
            __syncthreads();
        }
        if (threadIdx.x == 0) atomicAdd(&pullacc[bc], red[0]);
        __syncthreads();
    }
}

// ---------------------------------------------------------------------------
// Kernel 5: final scalar: mean_b( push[b] + 0.1 * pullacc[b] / (B*H*W) )
// ---------------------------------------------------------------------------
__global__ void final_kernel(const float* __restrict__ push,
                             const float* __restrict__ pullacc,
                             float* __restrict__ out)
{
    if (threadIdx.x == 0 && blockIdx.x == 0) {
        float s = 0.0f;
        for (int b = 0; b < BATCH; ++b)
            s += push[b] + 0.1f * (pullacc[b] / (float)(BATCH * NPIX));
        out[0] = s / (float)BATCH;
    }
}

// ---------------------------------------------------------------------------
// Workspace layout (floats):
//   [0,2048)      sums   [B][K][D]
//   [2048,2176)   counts [B][K]
//   [2176,4224)   cents  [B][K][D]
//   [4224,4228)   push   [B]
//   [4228,4232)   pullacc[B]
// ---------------------------------------------------------------------------
extern "C" void kernel_launch(void* const* d_in, const int* in_sizes, int n_in,
                              void* d_out, int out_size, void* d_ws, size_t ws_size,
                              hipStream_t stream) {
    const float* emb = (const float*)d_in[0];
    const int*   lab = (const int*)d_in[1];
    float* ws      = (float*)d_ws;
    float* sums    = ws;
    float* counts  = ws + 2048;
    float* cents   = ws + 2176;
    float* push    = ws + 4224;
    float* pullacc = ws + 4228;

    // zero all accumulators (cents/push overwritten, but cheap to clear all)
    (void)hipMemsetAsync(ws, 0, 4232 * sizeof(float), stream);

    dim3 cgrid(32, BATCH);
    centroid_kernel<<<cgrid, 256, 0, stream>>>(emb, lab, sums, counts);

    finalize_cents<<<(CENT_ELEMS + 255) / 256, 256, 0, stream>>>(
        sums, counts, cents);

    push_kernel<<<BATCH, 256, 0, stream>>>(cents, push);

    pull_kernel<<<1024, 256, 0, stream>>>(emb, lab, cents, pullacc);

    final_kernel<<<1, 64, 0, stream>>>(push, pullacc, (float*)d_out);
}